// Surface_smooth_Loss_46634754900367
// MI455X (gfx1250) — compile-verified
//
#include <hip/hip_runtime.h>
#include <hip/hip_bf16.h>

// Surface smoothness loss:  loss = ( sum_{b,f,c} |v1-v2|+|v1-v3|+|v3-v2| ) / (B*F*3)
// Strategy (MI455X / gfx1250):
//   stage1: one block per batch item. TDM (tensor_load_to_lds) DMAs the whole
//           82,680-byte vertex slab for this batch into LDS (fits in 320KB WGP
//           LDS), wave0 waits on TENSORcnt, then all 16 waves gather faces from
//           LDS and accumulate |diffs|. Deterministic wave+LDS reduction ->
//           per-block partial in d_ws.
//   stage2: fold 256 partials, scale by 1/(B*F*3), write scalar to d_out.

#define B_ 256
#define V_ 6890
#define F_ 13776
#define VERT_FLOATS (V_ * 3)            // 20670 floats = 82,680 bytes per batch
#define THREADS_1 512                   // 16 wave32s

typedef unsigned int u32x4 __attribute__((ext_vector_type(4)));
typedef int          i32x8 __attribute__((ext_vector_type(8)));
typedef int          i32x4 __attribute__((ext_vector_type(4)));

__device__ __forceinline__ float wave_reduce_f32(float v) {
    // wave32 butterfly reduction
    #pragma unroll
    for (int o = 16; o > 0; o >>= 1) v += __shfl_down(v, o, 32);
    return v;
}

__global__ void __launch_bounds__(THREADS_1)
surface_loss_stage1(const float* __restrict__ pv,
                    const int*   __restrict__ faces,
                    float*       __restrict__ partial) {
    extern __shared__ float smem[];            // [0,20670): verts  [20670,20686): wave sums
    float* verts = smem;
    float* wsum  = smem + VERT_FLOATS;

    const int b   = blockIdx.x;
    const int tid = threadIdx.x;

    // ---- Stage the batch's vertex slab into LDS via the Tensor Data Mover ----
    // D# describes a 2D tensor: dim0 = 20670 f32 elements, one row; the tile
    // covers the full row -> one contiguous 82,680B DMA into LDS offset 0
    // (dynamic-LDS base; no static LDS in this kernel).
    if ((tid >> 5) == 0) {                     // wave 0 issues the DMA (EXEC ignored by TDM)
        const unsigned long long gaddr =
            (unsigned long long)(const void*)(pv + (size_t)b * VERT_FLOATS);

        u32x4 g0;
        g0.x = 1u;                                              // count=1 (valid descriptor)
        g0.y = 0u;                                              // lds_addr = 0 (dynamic LDS base)
        g0.z = (unsigned)(gaddr & 0xFFFFFFFFull);               // global_addr[31:0]
        g0.w = (unsigned)((gaddr >> 32) & 0x1FFFFFFull)         // global_addr[56:32]
             | (2u << 30);                                      // type = 2 ("image")

        i32x8 g1;
        g1[0] = (int)(2u << 16);                                // data_size = 2 -> 4 bytes; no mask/pad/iter
        g1[1] = (int)(((unsigned)VERT_FLOATS & 0xFFFFu) << 16); // tensor_dim0[15:0]  (abar addr = 0)
        g1[2] = (int)(((unsigned)VERT_FLOATS >> 16)             // tensor_dim0[31:16] (=0)
             | (1u << 16));                                     // tensor_dim1[15:0] = 1
        g1[3] = (int)(((unsigned)VERT_FLOATS) << 16);           // tensor_dim1[31:16]=0 | tile_dim0 = 20670
        g1[4] = 1;                                              // tile_dim1 = 1, tile_dim2 = 0
        g1[5] = VERT_FLOATS;                                    // tensor_dim0_stride[31:0]
        g1[6] = 0;                                              // stride0 hi | stride1 lo
        g1[7] = 0;                                              // stride1 hi

        i32x4 g2 = {0, 0, 0, 0};                                // <=2D: groups 2/3 unused
        i32x4 g3 = {0, 0, 0, 0};
        i32x8 g4 = {0, 0, 0, 0, 0, 0, 0, 0};                    // 6-arg toolchain: extra group, unused

        __builtin_amdgcn_tensor_load_to_lds(g0, g1, g2, g3, g4, /*cpol=*/0);
        __builtin_amdgcn_s_wait_tensorcnt((short)0);            // s_wait_tensorcnt 0
    }
    __syncthreads();                                            // slab visible to all waves

    // ---- Gather faces from LDS and accumulate L1 edge lengths ----
    float acc = 0.0f;
    for (int f = tid; f < F_; f += THREADS_1) {
        // prefetch next face chunk into cache hierarchy (global_prefetch_b8)
        if (f + THREADS_1 < F_)
            __builtin_prefetch(faces + 3 * (f + THREADS_1), 0, 1);

        const int ia = faces[3 * f + 0];
        const int ib = faces[3 * f + 1];
        const int ic = faces[3 * f + 2];

        const float ax = verts[3 * ia + 0], ay = verts[3 * ia + 1], az = verts[3 * ia + 2];
        const float bx = verts[3 * ib + 0], by = verts[3 * ib + 1], bz = verts[3 * ib + 2];
        const float cx = verts[3 * ic + 0], cy = verts[3 * ic + 1], cz = verts[3 * ic + 2];

        acc += fabsf(ax - bx) + fabsf(ay - by) + fabsf(az - bz)   // |v1 - v2|
             + fabsf(ax - cx) + fabsf(ay - cy) + fabsf(az - cz)   // |v1 - v3|
             + fabsf(cx - bx) + fabsf(cy - by) + fabsf(cz - bz);  // |v3 - v2|
    }

    // ---- Deterministic block reduction: wave shuffle -> LDS -> wave 0 ----
    float w = wave_reduce_f32(acc);
    if ((tid & 31) == 0) wsum[tid >> 5] = w;   // 16 wave sums
    __syncthreads();
    if (tid < 32) {
        float v = (tid < (THREADS_1 / 32)) ? wsum[tid] : 0.0f;
        v = wave_reduce_f32(v);
        if (tid == 0) partial[b] = v;
    }
}

__global__ void __launch_bounds__(256)
surface_loss_stage2(const float* __restrict__ partial, float* __restrict__ out) {
    __shared__ float ws[8];
    const int tid = threadIdx.x;
    float v = partial[tid];                    // exactly 256 partials
    v = wave_reduce_f32(v);
    if ((tid & 31) == 0) ws[tid >> 5] = v;
    __syncthreads();
    if (tid < 32) {
        float s = (tid < 8) ? ws[tid] : 0.0f;
        s = wave_reduce_f32(s);
        if (tid == 0)
            out[0] = s * (1.0f / ((float)B_ * (float)F_ * 3.0f));
    }
}

extern "C" void kernel_launch(void* const* d_in, const int* in_sizes, int n_in,
                              void* d_out, int out_size, void* d_ws, size_t ws_size,
                              hipStream_t stream) {
    const float* pv    = (const float*)d_in[0];   // (B, V, 3) float32
    const int*   faces = (const int*)d_in[1];     // (F, 3) integer indices
    float* partial = (float*)d_ws;                // 256 floats of scratch
    float* outp    = (float*)d_out;

    const size_t shmem = (size_t)(VERT_FLOATS + 16) * sizeof(float); // ~82.7 KB dynamic LDS

    surface_loss_stage1<<<B_, THREADS_1, shmem, stream>>>(pv, faces, partial);
    surface_loss_stage2<<<1, 256, 0, stream>>>(partial, outp);
}